// QueryAndGroup_27341761806520
// MI455X (gfx1250) — compile-verified
//
#include <hip/hip_runtime.h>

// Problem constants (from reference setup_inputs)
#define BQ_B 4
#define BQ_N 8192
#define BQ_S 2048
#define BQ_C 64
#define BQ_K 32
#define BQ_R2 0.01f   // radius^2

typedef float v2f __attribute__((ext_vector_type(2)));
typedef float v8f __attribute__((ext_vector_type(8)));
typedef int   v4i __attribute__((ext_vector_type(4)));

#if __has_builtin(__builtin_amdgcn_global_load_async_to_lds_b128) && \
    __has_builtin(__builtin_amdgcn_s_wait_asynccnt)
#define USE_ASYNC_LDS 1
typedef __attribute__((address_space(1))) v4i gv4i_t;   // global b128 payload
typedef __attribute__((address_space(3))) v4i lv4i_t;   // LDS b128 payload
#endif

// ---- Kernel A: ball query via V_WMMA_F32_16X16X4_F32 ----------------------
// One wave (32 lanes) owns 32 centers (two 16-row WMMA tiles).
// Each step processes 16 points: 2 WMMAs compute the 32x16 dot-product tile,
// then d^2 < r^2  <=>  2*dot - (|x|^2 - r^2) > |c|^2  per element.
#define CPW 32                 // centers per wave
#define WPB 2                  // waves per block
#define TPB (WPB * 32)
#define CPB (WPB * CPW)        // centers per block

__global__ void __launch_bounds__(TPB)
ball_query_wmma(const float* __restrict__ xyz,
                const float* __restrict__ center,
                int* __restrict__ idx)
{
    extern __shared__ float s_xyz[];   // BQ_N * 3 floats = 96 KB (< 320 KB WGP LDS)

    const int b    = blockIdx.x / (BQ_S / CPB);
    const int blk  = blockIdx.x % (BQ_S / CPB);
    const int tid  = threadIdx.x;
    const int lane = tid & 31;
    const int wave = tid >> 5;

    // ---- stage this batch's xyz into LDS ----
    {
        const float* gbase = xyz + (size_t)b * BQ_N * 3;
#ifdef USE_ASYNC_LDS
        // CDNA5 async DMA: GLOBAL_LOAD_ASYNC_TO_LDS_B128 (ASYNCcnt), no VGPR
        // round trip for the 96 KB stage.
        for (int i = tid; i < (BQ_N * 3) / 4; i += TPB) {
            gv4i_t* gsrc = (gv4i_t*)((const char*)gbase + (size_t)i * 16);
            lv4i_t* ldst = (lv4i_t*)(unsigned)(size_t)((char*)s_xyz + (size_t)i * 16);
            __builtin_amdgcn_global_load_async_to_lds_b128(gsrc, ldst, 0, 0);
        }
        __builtin_amdgcn_s_wait_asynccnt(0);
#else
        const float4* g = (const float4*)gbase;
        float4*       s = (float4*)s_xyz;
        for (int i = tid; i < (BQ_N * 3) / 4; i += TPB) s[i] = g[i];
#endif
    }
    __syncthreads();

    const int     c0 = blk * CPB + wave * CPW;                 // first center of wave
    const float*  cc = center + ((size_t)b * BQ_S + c0) * 3;

    // lane r owns center c0+r for ordered emission
    const float mycx = cc[lane * 3 + 0];
    const float mycy = cc[lane * 3 + 1];
    const float mycz = cc[lane * 3 + 2];
    const float cn2own = mycx * mycx + mycy * mycy + mycz * mycz;

    // broadcast all 32 |c|^2 into wave-uniform (SGPR) values
    float cn2s[32];
#pragma unroll
    for (int j = 0; j < 32; ++j)
        cn2s[j] = __int_as_float(__builtin_amdgcn_readlane(__float_as_int(cn2own), j));

    // A matrices (16x4 f32 layout: lanes 0-15 hold K=0,1; lanes 16-31 hold K=2,3)
    const int m    = lane & 15;
    const int half = lane >> 4;
    v2f a0, a1;
    a0.x = cc[m * 3 + (half ? 2 : 0)];
    a0.y = half ? 0.0f : cc[m * 3 + 1];
    a1.x = cc[(m + 16) * 3 + (half ? 2 : 0)];
    a1.y = half ? 0.0f : cc[(m + 16) * 3 + 1];

    // ballot slot + half-word select for this lane's row mask
    const int src = ((lane >> 4) << 3) | (lane & 7);   // which ballot holds my row
    const int hh  = (lane >> 3) & 1;                   // low/high 16 bits

    int  cnt = 0;
    int  firstIdx = 0;
    int* __restrict__ myout = idx + ((size_t)b * BQ_S + c0 + lane) * BQ_K;

    // unroll x2: lets the scheduler overlap the next step's LDS loads with this
    // step's WMMA + epilogue instead of stalling on s_wait_dscnt each step
#pragma unroll 2
    for (int n0 = 0; n0 < BQ_N; n0 += 16) {
        // B matrix column (point) for this lane; same layout convention as A
        const float px = s_xyz[(n0 + m) * 3 + 0];
        const float py = s_xyz[(n0 + m) * 3 + 1];
        const float pz = s_xyz[(n0 + m) * 3 + 2];
        v2f bm;
        bm.x = half ? pz : px;
        bm.y = half ? 0.0f : py;
        const float tcol = (px * px + py * py + pz * pz) - BQ_R2;  // |x|^2 - r^2 per column

        v8f acc0 = {};
        v8f acc1 = {};
        acc0 = __builtin_amdgcn_wmma_f32_16x16x4_f32(false, a0, false, bm,
                                                     (short)0, acc0, false, false);
        acc1 = __builtin_amdgcn_wmma_f32_16x16x4_f32(false, a1, false, bm,
                                                     (short)0, acc1, false, false);

        // per-VGPR compare -> ballot; C/D layout: VGPR v = rows v (lanes 0-15)
        // and v+8 (lanes 16-31), column = lane&15
        unsigned bal[16];
#pragma unroll
        for (int v = 0; v < 8; ++v) {
            const float rc0 = half ? cn2s[v + 8] : cn2s[v];
            bal[v] = __builtin_amdgcn_ballot_w32(acc0[v] * 2.0f - tcol > rc0);
            const float rc1 = half ? cn2s[24 + v] : cn2s[16 + v];
            bal[8 + v] = __builtin_amdgcn_ballot_w32(acc1[v] * 2.0f - tcol > rc1);
        }

        // lane r selects its row's ballot half via an unrolled select chain
        // (bal[] values are wave-uniform SGPRs; ~3 VALU per slot)
        unsigned mask = 0u;
#pragma unroll
        for (int j = 0; j < 16; ++j) {
            const unsigned pick = hh ? (bal[j] >> 16) : (bal[j] & 0xFFFFu);
            mask = (src == j) ? pick : mask;
        }

        // ordered emission (sparse: ~0.4% density so mask is almost always 0)
        while (mask) {
            const int bit = __builtin_ctz(mask);
            mask &= mask - 1;
            const int p = n0 + bit;
            if (cnt == 0) firstIdx = p;
            if (cnt < BQ_K) myout[cnt] = p;
            ++cnt;
            if (cnt >= BQ_K) break;
        }

        // wave-uniform early exit when every center is full
        if (__builtin_amdgcn_ballot_w32(cnt < BQ_K) == 0u) break;
    }

    // pad remaining slots with first valid index (firstIdx==0 when none found)
    const int fillv = firstIdx;
    for (int k = (cnt < BQ_K ? cnt : BQ_K); k < BQ_K; ++k) myout[k] = fillv;
}

// ---- Kernel B: grouping/gather. lane == k (K=32 == wave32) ---------------
// Output: (B, 3+C, S, K); stores coalesced across lanes; feature gathers hit L2.
__global__ void __launch_bounds__(256)
group_gather(const float* __restrict__ xyz,
             const float* __restrict__ center,
             const float* __restrict__ feat,
             const int* __restrict__ idx,
             float* __restrict__ out)
{
    const int gw   = (int)blockIdx.x * 8 + (int)(threadIdx.x >> 5);  // (b,s) pair
    const int lane = threadIdx.x & 31;                               // k
    const int b = gw >> 11;            // / BQ_S
    const int s = gw & (BQ_S - 1);

    const int id = idx[((size_t)b * BQ_S + s) * BQ_K + lane];

    const float* p = xyz    + ((size_t)b * BQ_N + id) * 3;
    const float* c = center + ((size_t)b * BQ_S + s) * 3;

    const size_t SK = (size_t)BQ_S * BQ_K;
    float* o = out + (size_t)b * (BQ_C + 3) * SK + (size_t)s * BQ_K + lane;

    o[0]      = p[0] - c[0];
    o[SK]     = p[1] - c[1];
    o[2 * SK] = p[2] - c[2];

    const float* fb = feat + (size_t)b * BQ_C * BQ_N + id;
    o += 3 * SK;
#pragma unroll 8
    for (int ch = 0; ch < BQ_C; ++ch)
        o[(size_t)ch * SK] = fb[(size_t)ch * BQ_N];
}

extern "C" void kernel_launch(void* const* d_in, const int* in_sizes, int n_in,
                              void* d_out, int out_size, void* d_ws, size_t ws_size,
                              hipStream_t stream)
{
    (void)in_sizes; (void)n_in; (void)out_size; (void)ws_size;

    const float* xyz    = (const float*)d_in[0];   // (B,N,3)
    const float* center = (const float*)d_in[1];   // (B,S,3)
    const float* feat   = (const float*)d_in[2];   // (B,C,N)
    int*   idx = (int*)d_ws;                       // (B,S,K) int32 = 1 MB scratch
    float* out = (float*)d_out;                    // (B,3+C,S,K)

    const dim3 gridA(BQ_B * (BQ_S / CPB));                   // 128 blocks x 64 thr
    const size_t ldsA = (size_t)BQ_N * 3 * sizeof(float);    // 96 KB xyz stage
    ball_query_wmma<<<gridA, TPB, ldsA, stream>>>(xyz, center, idx);

    const dim3 gridB((BQ_B * BQ_S) / 8);                     // 1024 blocks x 256 thr
    group_gather<<<gridB, 256, 0, stream>>>(xyz, center, feat, idx, out);
}